// ResidualAttentionBlock_61529701482675
// MI455X (gfx1250) — compile-verified
//
#include <hip/hip_runtime.h>
#include <math.h>

typedef __attribute__((ext_vector_type(2))) float v2f;
typedef __attribute__((ext_vector_type(8))) float v8f;

#define BATCH  2
#define NTOK   1024
#define DMODEL 256
#define NHEAD  8
#define A2LEN  255
#define SPAD   256          // A2LEN padded to 16
#define NQ     2048         // NHEAD*DMODEL
#define WROW   257          // DMODEL+1 (bias in last column)
#define LDA    260          // LDS row stride for 16xK A tiles (bank-conflict free, 16B aligned rows)

// ---------------------------------------------------------------------------
// CDNA5 async global->LDS staging (ASYNCcnt path). Guarded: falls back to the
// known-good VGPR round-trip staging if the toolchain lacks the builtins.
// Builtin signature (from hipcc diagnostic): param0 = int __vector(4) * in
// the global AS; so: (v4i AS1* src, v4i AS3* dst, imm offset, imm cpol).
// ---------------------------------------------------------------------------
#if defined(__gfx1250__) &&                                              \
    __has_builtin(__builtin_amdgcn_global_load_async_to_lds_b128) &&     \
    __has_builtin(__builtin_amdgcn_s_wait_asynccnt)
#define USE_ASYNC_LDS 1
typedef int v4i_b128 __attribute__((vector_size(16)));
typedef __attribute__((address_space(1))) v4i_b128 gb128;
typedef __attribute__((address_space(3))) v4i_b128 lb128;
__device__ __forceinline__ void async_cp16(const float* g, float* l) {
    __builtin_amdgcn_global_load_async_to_lds_b128((gb128*)g, (lb128*)l, 0, 0);
}
__device__ __forceinline__ void async_wait0() {
    __builtin_amdgcn_s_wait_asynccnt(0);
}
#endif

// Stage a 16 x 256 f32 tile (row stride `srcld` floats) into s_a with LDS row
// stride LDA. Block has `nthr` threads. All addresses 16B-aligned -> B128.
__device__ __forceinline__ void stage_tile(float* s_a, const float* src,
                                           int srcld, int tid, int nthr) {
#ifdef USE_ASYNC_LDS
    for (int i = tid; i < 16 * (DMODEL / 4); i += nthr) {   // 1024 16B chunks
        int r = i >> 6, kq = (i & 63) * 4;
        async_cp16(src + r * srcld + kq, s_a + r * LDA + kq);
    }
    async_wait0();
#else
    for (int i = tid; i < 16 * DMODEL; i += nthr) {
        int r = i >> 8, k = i & 255;
        s_a[r * LDA + k] = src[r * srcld + k];
    }
#endif
}

// ---------------------------------------------------------------------------
// Kernel 0: out = x + fanout_bias (everywhere). Non-gathered tokens are done.
// ---------------------------------------------------------------------------
__global__ void k_prep(const float* __restrict__ x, const float* __restrict__ fw,
                       float* __restrict__ out) {
    int row = blockIdx.x;            // b*NTOK + t
    int c   = threadIdx.x;
    float bias = fw[c * WROW + DMODEL];
    out[row * DMODEL + c] = x[row * DMODEL + c] + bias;
}

// ---------------------------------------------------------------------------
// Kernel 1: gather xg[b][s][:] = x[b][a2a[s]][:]  (s==255 row zero-padded)
// ---------------------------------------------------------------------------
__global__ void k_gather(const float* __restrict__ x, const int* __restrict__ a2a,
                         float* __restrict__ xg) {
    int r = blockIdx.x;              // b*SPAD + s
    int b = r / SPAD, s = r % SPAD;
    int c = threadIdx.x;
    float v = 0.0f;
    if (s < A2LEN) v = x[(b * NTOK + a2a[s]) * DMODEL + c];
    xg[r * DMODEL + c] = v;
}

// ---------------------------------------------------------------------------
// Kernel 2: q/v projection GEMM via f32 WMMA 16x16x4.
// Out[s, o] = sum_i xg[b,s,i] * W[o,i] + W[o,256];  W = wq (o<2048) else wv.
// Block = 4 waves sharing one 16-row A tile in LDS; each wave owns a 16-col tile.
// ---------------------------------------------------------------------------
__global__ void k_proj(const float* __restrict__ xg,
                       const float* __restrict__ wq, const float* __restrict__ wv,
                       float* __restrict__ qg, float* __restrict__ vg) {
    __shared__ float s_a[16 * LDA];

    int bid = blockIdx.x;
    int ng  = bid & 63;              // 64 groups of 4 n-tiles -> 4096 cols
    int mt  = (bid >> 6) & 15;
    int b   = bid >> 10;
    int wave = threadIdx.x >> 5;
    int lane = threadIdx.x & 31;
    int m0 = mt * 16;
    int n0 = (ng * 4 + wave) * 16;   // global output column base (0..4080)

    stage_tile(s_a, xg + ((b * SPAD) + m0) * DMODEL, DMODEL, threadIdx.x, 128);
    __syncthreads();

    bool isV = (n0 >= NQ);
    const float* W = isV ? wv : wq;
    int  ncol = n0 + (lane & 15);
    int  wrow = ncol & (NQ - 1);     // column within q or v
    const float* wr = W + wrow * WROW;
    int  am = lane & 15;
    int  kk = (lane >> 4) * 2;

    v8f c = {0.f, 0.f, 0.f, 0.f, 0.f, 0.f, 0.f, 0.f};
    #pragma unroll 4
    for (int k0 = 0; k0 < DMODEL; k0 += 4) {
        v2f a = *reinterpret_cast<const v2f*>(s_a + am * LDA + k0 + kk);
        v2f bb;
        bb.x = wr[k0 + kk];
        bb.y = wr[k0 + kk + 1];
        c = __builtin_amdgcn_wmma_f32_16x16x4_f32(false, a, false, bb,
                                                  (short)0, c, false, false);
    }
    float bias = wr[DMODEL];
    float* OUT = isV ? vg : qg;
    #pragma unroll
    for (int r = 0; r < 8; ++r) {
        int row = r + ((lane >> 4) << 3);
        OUT[((b * SPAD) + m0 + row) * NQ + wrow] = c[r] + bias;
    }
}

// ---------------------------------------------------------------------------
// Kernel 3: L1-distance logits + masked softmax -> P[b][h][q][t]  (fp32)
// logit(q,t) = -1/16 * sum_d |qg[q,h,d] - xg[t,d]*wk[h,d]| ; phantom key logit 0
// Block handles (b, h, 16 consecutive q); thread tid == key t.
// ---------------------------------------------------------------------------
__global__ void k_attn(const float* __restrict__ xg, const float* __restrict__ qg,
                       const float* __restrict__ wk, float* __restrict__ P) {
    const int DC = 32, NCHUNK = DMODEL / DC;
    __shared__ float s_x[256 * (DC + 1)];     // keys chunk, stride 33
    __shared__ float s_q[16 * (DC + 1)];      // queries chunk
    __shared__ float s_wk[DC];
    __shared__ float s_l[16 * 257];           // logits, stride 257

    int bid = blockIdx.x;
    int qc = bid & 15;
    int h  = (bid >> 4) & 7;
    int b  = bid >> 7;
    int tid = threadIdx.x;                    // key index t (0..255)

    float acc[16];
    #pragma unroll
    for (int i = 0; i < 16; ++i) acc[i] = 0.0f;

    for (int dc = 0; dc < NCHUNK; ++dc) {
        int d0 = dc * DC;
        __syncthreads();
        for (int i = tid; i < 256 * DC; i += 256) {
            int t = i / DC, d = i % DC;
            s_x[t * (DC + 1) + d] = xg[((b * SPAD) + t) * DMODEL + d0 + d];
        }
        for (int i = tid; i < 16 * DC; i += 256) {
            int qi = i / DC, d = i % DC;
            int q = qc * 16 + qi;             // row 255 exists (padded)
            s_q[qi * (DC + 1) + d] = qg[((b * SPAD) + q) * NQ + h * DMODEL + d0 + d];
        }
        if (tid < DC) s_wk[tid] = wk[h * DMODEL + d0 + tid];
        __syncthreads();

        #pragma unroll 4
        for (int d = 0; d < DC; ++d) {
            float kv = s_x[tid * (DC + 1) + d] * s_wk[d];
            #pragma unroll
            for (int qi = 0; qi < 16; ++qi)
                acc[qi] += fabsf(s_q[qi * (DC + 1) + d] - kv);
        }
    }

    const float scale = 1.0f / 16.0f;         // 1/sqrt(256)
    __syncthreads();
    #pragma unroll
    for (int qi = 0; qi < 16; ++qi)
        s_l[qi * 257 + tid] = -scale * acc[qi];
    __syncthreads();

    // wave32 reductions: 8 waves x 2 queries each
    int wave = tid >> 5, lane = tid & 31;
    for (int qi = wave * 2; qi < wave * 2 + 2; ++qi) {
        float m = -INFINITY;
        #pragma unroll
        for (int j = 0; j < 8; ++j) {
            int t = lane + 32 * j;
            if (t < A2LEN) m = fmaxf(m, s_l[qi * 257 + t]);
        }
        #pragma unroll
        for (int off = 16; off > 0; off >>= 1)
            m = fmaxf(m, __shfl_xor(m, off, 32));
        m = fmaxf(m, 0.0f);                   // phantom key logit 0
        float s = 0.0f;
        #pragma unroll
        for (int j = 0; j < 8; ++j) {
            int t = lane + 32 * j;
            if (t < A2LEN) s += __expf(s_l[qi * 257 + t] - m);
        }
        #pragma unroll
        for (int off = 16; off > 0; off >>= 1)
            s += __shfl_xor(s, off, 32);
        float denom = s + __expf(-m);         // phantom key term (dropped output)
        int qglob = qc * 16 + qi;
        float* Prow = P + (((b * NHEAD + h) * SPAD) + qglob) * SPAD;
        #pragma unroll
        for (int j = 0; j < 8; ++j) {
            int t = lane + 32 * j;
            float p = 0.0f;
            if (qglob < A2LEN && t < A2LEN)
                p = __expf(s_l[qi * 257 + t] - m) / denom;
            Prow[t] = p;
        }
    }
}

// ---------------------------------------------------------------------------
// Kernel 4: bsum[b,q,w] = sum_h sum_t P[b,h,q,t] * vg[b,t,h*256+w]  (WMMA),
// fused with SiLU:  ys = bsum * sigmoid(1.702*bsum)
// ---------------------------------------------------------------------------
__global__ void k_attnv(const float* __restrict__ P, const float* __restrict__ vg,
                        float* __restrict__ ys) {
    __shared__ float s_a[16 * LDA];
    int bid = blockIdx.x;
    int ng = bid & 3;
    int mt = (bid >> 2) & 15;
    int b  = bid >> 6;
    int wave = threadIdx.x >> 5;
    int lane = threadIdx.x & 31;
    int m0 = mt * 16;                         // query tile
    int n0 = (ng * 4 + wave) * 16;            // w tile (0..255)
    int am = lane & 15;
    int kk = (lane >> 4) * 2;
    int ncol = n0 + (lane & 15);

    v8f c = {0.f, 0.f, 0.f, 0.f, 0.f, 0.f, 0.f, 0.f};
    for (int h = 0; h < NHEAD; ++h) {
        __syncthreads();
        stage_tile(s_a, P + (((b * NHEAD + h) * SPAD) + m0) * SPAD, SPAD,
                   threadIdx.x, 128);
        __syncthreads();
        const float* vb = vg + (size_t)(b * SPAD) * NQ + h * DMODEL + ncol;
        #pragma unroll 4
        for (int k0 = 0; k0 < SPAD; k0 += 4) {
            v2f a = *reinterpret_cast<const v2f*>(s_a + am * LDA + k0 + kk);
            int t = k0 + kk;
            v2f bb;
            bb.x = vb[(size_t)t * NQ];
            bb.y = vb[(size_t)(t + 1) * NQ];
            c = __builtin_amdgcn_wmma_f32_16x16x4_f32(false, a, false, bb,
                                                      (short)0, c, false, false);
        }
    }
    #pragma unroll
    for (int r = 0; r < 8; ++r) {
        int row = r + ((lane >> 4) << 3);
        float v = c[r];
        float sv = v / (1.0f + __expf(-1.702f * v));
        ys[((b * SPAD) + m0 + row) * DMODEL + ncol] = sv;
    }
}

// ---------------------------------------------------------------------------
// Kernel 5: fanout GEMM yo = ys @ Wf^T (WMMA) and scatter-add into out at a2a.
// (bias already added in k_prep; q==255 pad row is all-zero -> skipped)
// ---------------------------------------------------------------------------
__global__ void k_fanout(const float* __restrict__ ys, const float* __restrict__ fw,
                         const int* __restrict__ a2a, float* __restrict__ out) {
    __shared__ float s_a[16 * LDA];
    int bid = blockIdx.x;
    int ng = bid & 3;
    int mt = (bid >> 2) & 15;
    int b  = bid >> 6;
    int wave = threadIdx.x >> 5;
    int lane = threadIdx.x & 31;
    int m0 = mt * 16;
    int n0 = (ng * 4 + wave) * 16;
    int am = lane & 15;
    int kk = (lane >> 4) * 2;
    int ncol = n0 + (lane & 15);

    stage_tile(s_a, ys + ((b * SPAD) + m0) * DMODEL, DMODEL, threadIdx.x, 128);
    __syncthreads();

    const float* wr = fw + ncol * WROW;
    v8f c = {0.f, 0.f, 0.f, 0.f, 0.f, 0.f, 0.f, 0.f};
    #pragma unroll 4
    for (int k0 = 0; k0 < DMODEL; k0 += 4) {
        v2f a = *reinterpret_cast<const v2f*>(s_a + am * LDA + k0 + kk);
        v2f bb;
        bb.x = wr[k0 + kk];
        bb.y = wr[k0 + kk + 1];
        c = __builtin_amdgcn_wmma_f32_16x16x4_f32(false, a, false, bb,
                                                  (short)0, c, false, false);
    }
    #pragma unroll
    for (int r = 0; r < 8; ++r) {
        int row = r + ((lane >> 4) << 3);
        int q = m0 + row;
        if (q < A2LEN) {
            int tok = a2a[q];
            float* o = out + ((size_t)(b * NTOK + tok)) * DMODEL + ncol;
            *o += c[r];                       // unique (b,tok,col) per lane: no race
        }
    }
}

// ---------------------------------------------------------------------------
extern "C" void kernel_launch(void* const* d_in, const int* in_sizes, int n_in,
                              void* d_out, int out_size, void* d_ws, size_t ws_size,
                              hipStream_t stream) {
    const float* x   = (const float*)d_in[0];
    const float* wq  = (const float*)d_in[1];
    const float* wv  = (const float*)d_in[2];
    const float* wk  = (const float*)d_in[3];
    const float* fw  = (const float*)d_in[4];
    const int*   a2a = (const int*)d_in[5];
    float* out = (float*)d_out;

    float* ws = (float*)d_ws;
    float* xg = ws;                                    // 2*256*256
    float* qg = xg + BATCH * SPAD * DMODEL;            // 2*256*2048
    float* vg = qg + BATCH * SPAD * NQ;                // 2*256*2048
    float* P  = vg + BATCH * SPAD * NQ;                // 2*8*256*256
    float* ys = P  + BATCH * NHEAD * SPAD * SPAD;      // 2*256*256

    k_prep  <<<BATCH * NTOK,       DMODEL, 0, stream>>>(x, fw, out);
    k_gather<<<BATCH * SPAD,       DMODEL, 0, stream>>>(x, a2a, xg);
    k_proj  <<<BATCH * 16 * 64,    128,    0, stream>>>(xg, wq, wv, qg, vg);
    k_attn  <<<BATCH * NHEAD * 16, 256,    0, stream>>>(xg, qg, wk, P);
    k_attnv <<<BATCH * 16 * 4,     128,    0, stream>>>(P, vg, ys);
    k_fanout<<<BATCH * 16 * 4,     128,    0, stream>>>(ys, fw, a2a, out);
}